// LightGCN_61117384622812
// MI455X (gfx1250) — compile-verified
//
#include <hip/hip_runtime.h>
#include <hip/hip_bf16.h>

#define LG_D 128

typedef float v2f __attribute__((ext_vector_type(2)));
typedef float v8f __attribute__((ext_vector_type(8)));

#if defined(__has_builtin)
#if __has_builtin(__builtin_amdgcn_wmma_f32_16x16x4_f32)
#define HAVE_WMMA_F32X4 1
#endif
#endif

// ---------------------------------------------------------------------------
// Kernel 1: re-initialize workspace every launch (graph replay safe).
// map[i] = -1 for all N nodes, e1c zeroed, slot counter zeroed.
// ---------------------------------------------------------------------------
__global__ void lg_init(int* __restrict__ map, float* __restrict__ e1c,
                        int* __restrict__ counter, int N, int slotsD) {
    int i = blockIdx.x * blockDim.x + threadIdx.x;
    if (i < N) map[i] = -1;
    if (i < slotsD) e1c[i] = 0.0f;
    if (i == 0) *counter = 0;
}

// ---------------------------------------------------------------------------
// Kernel 2: build node -> compact-slot map for the sampled users/items.
// ---------------------------------------------------------------------------
__global__ void lg_build_map(const int* __restrict__ uidx, const int* __restrict__ iidx,
                             int* __restrict__ map, int* __restrict__ counter,
                             int Bn, int U) {
    int b = blockIdx.x * blockDim.x + threadIdx.x;
    if (b >= 2 * Bn) return;
    int node = (b < Bn) ? uidx[b] : (U + iidx[b - Bn]);
    if (atomicCAS(&map[node], -1, -2) == -1) {
        int slot = atomicAdd(counter, 1);
        map[node] = slot;   // visible to later kernels at kernel boundary
    }
}

// ---------------------------------------------------------------------------
// Kernel 3: filtered COO scatter. One wave32 per edge (grid-stride).
// Only edges whose row is a sampled node contribute (~10% of 6.4M edges).
// Each lane loads float4 (global_load_b128, coalesced 512B/row) and does
// 4x global_atomic_add_f32 into the compact accumulation row.
// ---------------------------------------------------------------------------
__global__ void lg_scatter(const float* __restrict__ user_emb,
                           const float* __restrict__ item_emb,
                           const float* __restrict__ vals,
                           const int* __restrict__ rows,
                           const int* __restrict__ cols,
                           const int* __restrict__ map,
                           float* __restrict__ e1c,
                           int nnz, int U) {
    int lane = threadIdx.x & 31;
    long long gtid   = (long long)blockIdx.x * blockDim.x + threadIdx.x;
    long long wid    = gtid >> 5;
    long long nwaves = ((long long)gridDim.x * blockDim.x) >> 5;

    for (long long e = wid; e < nnz; e += nwaves) {
        int row  = rows[e];         // wave-uniform: all lanes same edge
        int slot = map[row];
        if (slot < 0) continue;     // uniform branch, no divergence
        int   col = cols[e];
        float v   = vals[e];
        const float* src = (col < U) ? (user_emb + (long long)col * LG_D)
                                     : (item_emb + (long long)(col - U) * LG_D);
        const float4* src4 = reinterpret_cast<const float4*>(src);
        float4 x = src4[lane];                       // 128B/lane-group coalesced
        float* dst = e1c + (long long)slot * LG_D + lane * 4;
        atomicAdd(dst + 0, v * x.x);
        atomicAdd(dst + 1, v * x.y);
        atomicAdd(dst + 2, v * x.z);
        atomicAdd(dst + 3, v * x.w);
    }
}

// ---------------------------------------------------------------------------
// Kernel 4: gather final = 0.5*(e0 + e1) rows for each batch entry into
// dense [B,128] buffers (one block of 128 threads per row; coalesced).
// blocks [0,B)   -> user rows into Uf
// blocks [B,2B)  -> item rows into Tf
// ---------------------------------------------------------------------------
__global__ void lg_gather(const float* __restrict__ user_emb,
                          const float* __restrict__ item_emb,
                          const float* __restrict__ e1c,
                          const int* __restrict__ map,
                          const int* __restrict__ uidx,
                          const int* __restrict__ iidx,
                          float* __restrict__ Uf, float* __restrict__ Tf,
                          int Bn, int U) {
    int b = blockIdx.x;
    int d = threadIdx.x;     // 0..127
    if (b < Bn) {
        int u    = uidx[b];
        int slot = map[u];
        float e1 = (slot >= 0) ? e1c[(long long)slot * LG_D + d] : 0.0f;
        Uf[(long long)b * LG_D + d] = 0.5f * (user_emb[(long long)u * LG_D + d] + e1);
    } else {
        int bb   = b - Bn;
        int it   = iidx[bb];
        int slot = map[U + it];
        float e1 = (slot >= 0) ? e1c[(long long)slot * LG_D + d] : 0.0f;
        Tf[(long long)bb * LG_D + d] = 0.5f * (item_emb[(long long)it * LG_D + d] + e1);
    }
}

// ---------------------------------------------------------------------------
// Kernel 5: batched dot via WMMA. Each wave32 owns 16 (user,item) pairs and
// computes C = U_tile(16x128) @ T_tile^T(128x16) as 32 chained
// v_wmma_f32_16x16x4_f32 ops, then extracts the diagonal C[m][m].
//
// Fragment layouts (ISA 7.12.2, 32-bit 16x4 A):
//   lanes 0-15 : M=lane,    vgpr0=K0, vgpr1=K1
//   lanes 16-31: M=lane-16, vgpr0=K2, vgpr1=K3
// B (4x16) mirrors A with M<->N. C (16x16 f32): lane n (0-15) vgpr r -> M=r;
// lane n+16 vgpr r -> M=r+8. Diagonal: m<8 at (lane m, c[m]); m>=8 at
// (lane m+16, c[m-8]).
// ---------------------------------------------------------------------------
__global__ void lg_dot_wmma(const float* __restrict__ Uf,
                            const float* __restrict__ Tf,
                            float* __restrict__ out, int Bn) {
    long long gtid = (long long)blockIdx.x * blockDim.x + threadIdx.x;
    int  lane = (int)(gtid & 31);
    int  base = (int)((gtid >> 5) * 16);        // wave-uniform
    if (base >= Bn) return;                     // whole wave exits together

    int mrow  = lane & 15;
    int khalf = (lane >> 4) * 2;                // 0 or 2
    const float* urow = Uf + (long long)(base + mrow) * LG_D;
    const float* trow = Tf + (long long)(base + mrow) * LG_D;

#if defined(HAVE_WMMA_F32X4)
    v8f c = {};
#pragma unroll
    for (int d0 = 0; d0 < LG_D; d0 += 4) {
        v2f a, b;
        a.x = urow[d0 + khalf + 0];
        a.y = urow[d0 + khalf + 1];
        b.x = trow[d0 + khalf + 0];
        b.y = trow[d0 + khalf + 1];
        c = __builtin_amdgcn_wmma_f32_16x16x4_f32(
                /*neg_a=*/false, a, /*neg_b=*/false, b,
                /*c_mod=*/(short)0, c, /*reuse_a=*/false, /*reuse_b=*/false);
    }
#pragma unroll
    for (int r = 0; r < 8; ++r) {
        if (lane == r)      out[base + r]     = c[r];   // M = r
        if (lane == r + 24) out[base + r + 8] = c[r];   // M = r + 8
    }
#else
    // VALU fallback: lanes 0-15 each compute one full dot.
    float acc = 0.0f;
#pragma unroll 8
    for (int d = 0; d < LG_D; ++d) acc += urow[d] * trow[d];
    if (lane < 16) out[base + lane] = acc;
#endif
}

// ---------------------------------------------------------------------------
extern "C" void kernel_launch(void* const* d_in, const int* in_sizes, int n_in,
                              void* d_out, int out_size, void* d_ws, size_t ws_size,
                              hipStream_t stream) {
    const float* user_emb = (const float*)d_in[0];
    const float* item_emb = (const float*)d_in[1];
    const float* adj_vals = (const float*)d_in[2];
    const int*   adj_rows = (const int*)d_in[3];
    const int*   adj_cols = (const int*)d_in[4];
    const int*   uidx     = (const int*)d_in[5];
    const int*   iidx     = (const int*)d_in[6];
    float*       out      = (float*)d_out;

    const int U   = in_sizes[0] / LG_D;   // 100000
    const int I   = in_sizes[1] / LG_D;   // 200000
    const int nnz = in_sizes[2];          // 6.4M
    const int Bn  = in_sizes[5];          // 16384
    const int N   = U + I;
    const int slots  = 2 * Bn;            // upper bound on unique sampled nodes
    const int slotsD = slots * LG_D;

    // Workspace carve-up (~35 MB total).
    char*  ws  = (char*)d_ws;
    size_t off = 0;
    auto align256 = [](size_t x) { return (x + 255) & ~(size_t)255; };
    int*   map     = (int*)(ws + off);  off = align256(off + (size_t)N * sizeof(int));
    int*   counter = (int*)(ws + off);  off = align256(off + 256);
    float* e1c     = (float*)(ws + off); off = align256(off + (size_t)slotsD * sizeof(float));
    float* Uf      = (float*)(ws + off); off = align256(off + (size_t)Bn * LG_D * sizeof(float));
    float* Tf      = (float*)(ws + off); off = align256(off + (size_t)Bn * LG_D * sizeof(float));

    // 1) init
    {
        int n = (N > slotsD) ? N : slotsD;
        lg_init<<<(n + 255) / 256, 256, 0, stream>>>(map, e1c, counter, N, slotsD);
    }
    // 2) slot map
    lg_build_map<<<(2 * Bn + 255) / 256, 256, 0, stream>>>(uidx, iidx, map, counter, Bn, U);
    // 3) filtered edge scatter (grid-stride, one wave per edge)
    lg_scatter<<<2048, 256, 0, stream>>>(user_emb, item_emb, adj_vals,
                                         adj_rows, adj_cols, map, e1c, nnz, U);
    // 4) gather final rows for the batch
    lg_gather<<<2 * Bn, LG_D, 0, stream>>>(user_emb, item_emb, e1c, map,
                                           uidx, iidx, Uf, Tf, Bn, U);
    // 5) WMMA diagonal dot: 16 pairs per wave
    {
        int waves   = (Bn + 15) / 16;
        int threads = waves * 32;
        lg_dot_wmma<<<(threads + 255) / 256, 256, 0, stream>>>(Uf, Tf, out, Bn);
    }
}